// TradeFlowRGCN_10668698764070
// MI455X (gfx1250) — compile-verified
//
#include <hip/hip_runtime.h>
#include <math.h>

#define N_NODES 50000
#define N_EDGES 800000
#define F_IN 3
#define HDIM 64
#define NREL 3
#define EAD 6

typedef float v2f __attribute__((ext_vector_type(2)));
typedef float v8f __attribute__((ext_vector_type(8)));

// f32 WMMA: D(16x16) = A(16x4) * B(4x16) + C, full f32 precision (matches reference dtype)
__device__ __forceinline__ v8f wmma4(v2f a, v2f b, v8f c) {
  return __builtin_amdgcn_wmma_f32_16x16x4_f32(false, a, false, b, (short)0, c, false, false);
}

__global__ void k_zero(float* __restrict__ p, int n) {
  int i = blockIdx.x * blockDim.x + threadIdx.x;
  int stride = gridDim.x * blockDim.x;
  for (; i < n; i += stride) p[i] = 0.f;
}

__global__ void k_edge_type(const float* __restrict__ ea, const int* __restrict__ dst,
                            int* __restrict__ et, float* __restrict__ cnt) {
  int e = blockIdx.x * blockDim.x + threadIdx.x;
  if (e >= N_EDGES) return;
  float dist = ea[e * EAD];
  const float q1 = 8.51739317f;  // log1p(5000)
  const float q2 = 9.21044037f;  // log1p(10000)
  int t = (dist > q1 ? 1 : 0) + (dist > q2 ? 1 : 0);
  et[e] = t;
  __hip_atomic_fetch_add(&cnt[t * N_NODES + dst[e]], 1.0f,
                         __ATOMIC_RELAXED, __HIP_MEMORY_SCOPE_AGENT);
}

__global__ void k_inv(const float* __restrict__ cnt, float* __restrict__ inv) {
  int i = blockIdx.x * blockDim.x + threadIdx.x;
  if (i < NREL * N_NODES) inv[i] = 1.0f / fmaxf(cnt[i], 1.0f);
}

// scatter-add gathered source features into per-relation accumulators (L2-resident)
template <int F>
__global__ void k_scatter(const int* __restrict__ src, const int* __restrict__ dst,
                          const int* __restrict__ et, const float* __restrict__ xin,
                          float* __restrict__ s) {
  int i = blockIdx.x * blockDim.x + threadIdx.x;
  if (i >= N_EDGES * F) return;
  int e = i / F;
  int k = i - e * F;
  int t = et[e];
  float v = xin[src[e] * F + k];
  __hip_atomic_fetch_add(&s[(t * N_NODES + dst[e]) * F + k], v,
                         __ATOMIC_RELAXED, __HIP_MEMORY_SCOPE_AGENT);
}

// One RGCN layer: out = LN( relu( [x|s0*i0|s1*i1|s2*i2] @ [root;W0;W1;W2] + b ) + residual )
// Block: 128 threads = 4 waves, 64 nodes. Each wave does a 16x64 output tile via
// f32 WMMA 16x16x4. Weights stored PAIR-INTERLEAVED in LDS ([k/2][col][k%2]) so each
// B fragment is one aligned ds_load_b64 into an even VGPR pair.
template <int F, int FPAD, int SHIFT>
__global__ void k_rgcn(const float* __restrict__ xin, const float* __restrict__ s,
                       const float* __restrict__ inv, const float* __restrict__ root,
                       const float* __restrict__ W, const float* __restrict__ bias,
                       const float* __restrict__ lng, const float* __restrict__ lnb,
                       int resmode, const float* __restrict__ resw_or_prev,
                       const float* __restrict__ resb, float* __restrict__ out) {
  constexpr int KT = 4 * FPAD;
  extern __shared__ float smem[];
  float* Bm = smem;              // KT*64, pair-interleaved: Bm[((k>>1)*64 + c)*2 + (k&1)]
  float* stag = smem + KT * 64;  // 64*64 output staging for LayerNorm
  const int tid = threadIdx.x;

  for (int i = tid; i < KT * 64; i += 128) {
    int k = i >> 6, c = i & 63;
    int seg = k >> SHIFT, lk = k & (FPAD - 1);
    float v = 0.f;
    if (lk < F) v = (seg == 0) ? root[lk * HDIM + c] : W[((seg - 1) * F + lk) * HDIM + c];
    Bm[((k >> 1) * 64 + c) * 2 + (k & 1)] = v;
  }
  __syncthreads();

  const int lane = tid & 31, wv = tid >> 5;
  const int m0 = blockIdx.x * 64 + wv * 16;
  if (m0 < N_NODES) {  // N % 16 == 0: whole wave-tile valid or not
    const int row = m0 + (lane & 15);
    const int koff = (lane < 16) ? 0 : 2;  // ISA f32 A layout: lanes 0-15 K=0,1; 16-31 K=2,3
    const int col0 = lane & 15;
    v8f acc[4] = {{}, {}, {}, {}};

    auto stepd = [&](v2f a, int krow) {  // krow: even B-row (koff folded in)
#pragma unroll
      for (int nt = 0; nt < 4; ++nt) {
        v2f b = *(const v2f*)(Bm + ((krow >> 1) * 64 + nt * 16 + col0) * 2);
        acc[nt] = wmma4(a, b, acc[nt]);
      }
    };

    if constexpr (F == HDIM) {
      // segment 0: x @ root
      {
        const float* p = xin + row * HDIM;
#pragma unroll 2
        for (int kb = 0; kb < 16; ++kb) {
          int kl = kb * 4 + koff;
          v2f a = *(const v2f*)(p + kl);
          stepd(a, kl);
        }
      }
      // segments 1..3: (s_r * inv_r) @ W_r
      for (int r = 0; r < 3; ++r) {
        int idx = r * N_NODES + row;
        const float* p = s + idx * HDIM;
        float scale = inv[idx];
#pragma unroll 2
        for (int kb = 0; kb < 16; ++kb) {
          int kl = kb * 4 + koff;
          v2f a = *(const v2f*)(p + kl);
          a.x *= scale;
          a.y *= scale;
          stepd(a, (r + 1) * HDIM + kl);
        }
      }
    } else {
      // F == 3, each segment padded to K=4: one k-block per segment.
      {
        const float* p = xin + row * 3;
        v2f a;
        if (lane < 16) { a.x = p[0]; a.y = p[1]; }
        else           { a.x = p[2]; a.y = 0.f; }
        stepd(a, koff);
      }
      for (int r = 0; r < 3; ++r) {
        int idx = r * N_NODES + row;
        const float* p = s + idx * 3;
        float scale = inv[idx];
        v2f a;
        if (lane < 16) { a.x = p[0] * scale; a.y = p[1] * scale; }
        else           { a.x = p[2] * scale; a.y = 0.f; }
        stepd(a, (r + 1) * 4 + koff);
      }
    }

#pragma unroll
    for (int nt = 0; nt < 4; ++nt)
#pragma unroll
      for (int vv = 0; vv < 8; ++vv) {
        int r = wv * 16 + vv + ((lane < 16) ? 0 : 8);  // D layout: M=v / v+8
        stag[r * 64 + nt * 16 + col0] = acc[nt][vv];
      }
  }
  __syncthreads();

  if (tid < 64) {
    int n = blockIdx.x * 64 + tid;
    if (n < N_NODES) {
      float vrow[64];
      float mu = 0.f;
      for (int c = 0; c < 64; ++c) {
        float h = fmaxf(stag[tid * 64 + c] + bias[c], 0.f);
        float res;
        if (resmode == 0) {
          res = resb[c];
#pragma unroll
          for (int k = 0; k < F_IN; ++k) res += xin[n * F + k] * resw_or_prev[k * HDIM + c];
        } else {
          res = resw_or_prev[n * HDIM + c];
        }
        h += res;
        vrow[c] = h;
        mu += h;
      }
      mu *= (1.f / 64.f);
      float var = 0.f;
      for (int c = 0; c < 64; ++c) { float d = vrow[c] - mu; var += d * d; }
      var *= (1.f / 64.f);
      float rstd = rsqrtf(var + 1e-5f);
      for (int c = 0; c < 64; ++c)
        out[n * HDIM + c] = (vrow[c] - mu) * rstd * lng[c] + lnb[c];
    }
  }
}

// Decoder: 128 edges / block (256 thr = 8 waves). Stage h3[src], h3[dst], ea, dec_w1
// in LDS. Fused inner loop: load (hs,hd) pair once, issue 8 WMMAs (hs, hd, |hs-hd|,
// hs*hd against 4 W1 row-blocks). W1 pair-interleaved -> one ds_load_b64 per fragment.
// Branch-free tail covers the 6 edge-attr columns with K padded 262 -> 264.
#define DEC_SMEM_FLOATS (264 * 32 + 128 * 64 + 128 * 64 + 128 * 6 + 128 * 32 + 512 + 16 + 32 + 16)
__global__ void k_decoder(const int* __restrict__ src, const int* __restrict__ dst,
                          const float* __restrict__ h3, const float* __restrict__ ea,
                          const float* __restrict__ w1, const float* __restrict__ b1,
                          const float* __restrict__ w2, const float* __restrict__ b2,
                          const float* __restrict__ w3, const float* __restrict__ b3,
                          float* __restrict__ out) {
  extern __shared__ float smem[];
  float* W1 = smem;            // 264*32 pair-interleaved: W1[((k>>1)*32 + c)*2 + (k&1)]
  float* HS = W1 + 264 * 32;   // 128*64
  float* HD = HS + 128 * 64;   // 128*64
  float* EAs = HD + 128 * 64;  // 128*6
  float* Z1 = EAs + 128 * 6;   // 128*32
  float* W2s = Z1 + 128 * 32;  // 32*16
  float* W3s = W2s + 512;      // 16
  float* B1s = W3s + 16;       // 32
  float* B2s = B1s + 32;       // 16
  const int tid = threadIdx.x;
  const int eb0 = blockIdx.x * 128;

  for (int i = tid; i < 264 * 32; i += 256) {
    int k = i >> 5, c = i & 31;
    W1[((k >> 1) * 32 + c) * 2 + (k & 1)] = (k < 262) ? w1[k * 32 + c] : 0.f;
  }
  for (int i = tid; i < 128 * 64; i += 256) {
    int le = i >> 6, c = i & 63;
    int e = eb0 + le;
    HS[i] = h3[src[e] * HDIM + c];
    HD[i] = h3[dst[e] * HDIM + c];
  }
  for (int i = tid; i < 128 * EAD; i += 256) EAs[i] = ea[eb0 * EAD + i];
  if (tid < 512) W2s[tid] = w2[tid];
  if (tid < 16) W3s[tid] = w3[tid];
  if (tid < 32) B1s[tid] = b1[tid];
  if (tid < 16) B2s[tid] = b2[tid];
  __syncthreads();

  const int lane = tid & 31, wv = tid >> 5;
  const int lrow = wv * 16 + (lane & 15);
  const int koff = (lane < 16) ? 0 : 2;
  const int col0 = lane & 15;
  const int hbase = lrow * 64;
  v8f acc0 = {}, acc1 = {};

  auto step = [&](v2f a, int krow) {  // krow even
    v2f b0 = *(const v2f*)(W1 + ((krow >> 1) * 32 + col0) * 2);
    acc0 = wmma4(a, b0, acc0);
    v2f b1v = *(const v2f*)(W1 + ((krow >> 1) * 32 + 16 + col0) * 2);
    acc1 = wmma4(a, b1v, acc1);
  };

  for (int kb = 0; kb < 16; ++kb) {
    int kl = kb * 4 + koff;
    v2f p = *(const v2f*)(HS + hbase + kl);  // hs pair
    v2f q = *(const v2f*)(HD + hbase + kl);  // hd pair
    v2f a;
    a.x = p.x;              a.y = p.y;              step(a, kl);          // hs
    a.x = q.x;              a.y = q.y;              step(a, 64 + kl);     // hd
    a.x = fabsf(p.x - q.x); a.y = fabsf(p.y - q.y); step(a, 128 + kl);    // |hs-hd|
    a.x = p.x * q.x;        a.y = p.y * q.y;        step(a, 192 + kl);    // hs*hd
  }
  {  // edge-attr tail: kk in [256,264), valid up to 261
    v2f a;
    a.x = EAs[lrow * EAD + koff];
    a.y = EAs[lrow * EAD + koff + 1];
    step(a, 256 + koff);
    a.x = (lane < 16) ? EAs[lrow * EAD + 4] : 0.f;
    a.y = (lane < 16) ? EAs[lrow * EAD + 5] : 0.f;
    step(a, 260 + koff);
  }

#pragma unroll
  for (int vv = 0; vv < 8; ++vv) {
    int r = wv * 16 + vv + ((lane < 16) ? 0 : 8);
    Z1[r * 32 + col0]      = fmaxf(acc0[vv] + B1s[col0], 0.f);
    Z1[r * 32 + 16 + col0] = fmaxf(acc1[vv] + B1s[16 + col0], 0.f);
  }
  __syncthreads();

  if (tid < 128) {
    int e = eb0 + tid;
    float zacc[16];
#pragma unroll
    for (int k2 = 0; k2 < 16; ++k2) zacc[k2] = B2s[k2];
    for (int j = 0; j < 32; ++j) {
      float zj = Z1[tid * 32 + j];
#pragma unroll
      for (int k2 = 0; k2 < 16; ++k2) zacc[k2] += zj * W2s[j * 16 + k2];
    }
    float o = b3[0];
#pragma unroll
    for (int k2 = 0; k2 < 16; ++k2) o += fmaxf(zacc[k2], 0.f) * W3s[k2];
    out[e] = o;
  }
}

extern "C" void kernel_launch(void* const* d_in, const int* in_sizes, int n_in,
                              void* d_out, int out_size, void* d_ws, size_t ws_size,
                              hipStream_t stream) {
  (void)in_sizes; (void)n_in; (void)out_size; (void)ws_size;
  const float* x    = (const float*)d_in[0];
  const int*   ei   = (const int*)d_in[1];
  const float* ea   = (const float*)d_in[2];
  const float* w0   = (const float*)d_in[3];
  const float* r0   = (const float*)d_in[4];
  const float* b0   = (const float*)d_in[5];
  const float* w1   = (const float*)d_in[6];
  const float* r1   = (const float*)d_in[7];
  const float* b1   = (const float*)d_in[8];
  const float* w2   = (const float*)d_in[9];
  const float* r2   = (const float*)d_in[10];
  const float* b2   = (const float*)d_in[11];
  const float* lng0 = (const float*)d_in[12];
  const float* lnb0 = (const float*)d_in[13];
  const float* lng1 = (const float*)d_in[14];
  const float* lnb1 = (const float*)d_in[15];
  const float* lng2 = (const float*)d_in[16];
  const float* lnb2 = (const float*)d_in[17];
  const float* resw = (const float*)d_in[18];
  const float* resb = (const float*)d_in[19];
  const float* dw1  = (const float*)d_in[20];
  const float* db1  = (const float*)d_in[21];
  const float* dw2  = (const float*)d_in[22];
  const float* db2  = (const float*)d_in[23];
  const float* dw3  = (const float*)d_in[24];
  const float* db3  = (const float*)d_in[25];
  const int* src = ei;
  const int* dstp = ei + N_EDGES;
  float* outp = (float*)d_out;

  char* ws = (char*)d_ws;
  size_t off = 0;
  auto wsalloc = [&](size_t bytes) {
    void* p = ws + off;
    off += (bytes + 255) & ~(size_t)255;
    return p;
  };
  int*   et   = (int*)  wsalloc((size_t)N_EDGES * 4);
  float* cnt  = (float*)wsalloc((size_t)NREL * N_NODES * 4);
  float* inv  = (float*)wsalloc((size_t)NREL * N_NODES * 4);
  float* sbuf = (float*)wsalloc((size_t)NREL * N_NODES * HDIM * 4);
  float* hA   = (float*)wsalloc((size_t)N_NODES * HDIM * 4);
  float* hB   = (float*)wsalloc((size_t)N_NODES * HDIM * 4);

  const int rgcnSmem = (256 * 64 + 64 * 64) * 4;   // 80 KB (dense layers)
  const int rgcn0Smem = (16 * 64 + 64 * 64) * 4;   // 20 KB (layer 0)
  const int decSmem = DEC_SMEM_FLOATS * 4;         // ~118 KB (of 320 KB/WGP)
  hipFuncSetAttribute((const void*)k_rgcn<HDIM, 64, 6>,
                      hipFuncAttributeMaxDynamicSharedMemorySize, rgcnSmem);
  hipFuncSetAttribute((const void*)k_decoder,
                      hipFuncAttributeMaxDynamicSharedMemorySize, decSmem);

  k_zero<<<1024, 256, 0, stream>>>(cnt, NREL * N_NODES);
  k_edge_type<<<(N_EDGES + 255) / 256, 256, 0, stream>>>(ea, dstp, et, cnt);
  k_inv<<<(NREL * N_NODES + 255) / 256, 256, 0, stream>>>(cnt, inv);

  const int rgcnBlocks = (N_NODES + 63) / 64;

  // Layer 0 (F = 3, K padded to 4 per segment)
  k_zero<<<1024, 256, 0, stream>>>(sbuf, NREL * N_NODES * F_IN);
  k_scatter<F_IN><<<(N_EDGES * F_IN + 255) / 256, 256, 0, stream>>>(src, dstp, et, x, sbuf);
  k_rgcn<F_IN, 4, 2><<<rgcnBlocks, 128, rgcn0Smem, stream>>>(
      x, sbuf, inv, r0, w0, b0, lng0, lnb0, /*resmode=*/0, resw, resb, hA);

  // Layer 1
  k_zero<<<2048, 256, 0, stream>>>(sbuf, NREL * N_NODES * HDIM);
  k_scatter<HDIM><<<(N_EDGES * HDIM + 255) / 256, 256, 0, stream>>>(src, dstp, et, hA, sbuf);
  k_rgcn<HDIM, 64, 6><<<rgcnBlocks, 128, rgcnSmem, stream>>>(
      hA, sbuf, inv, r1, w1, b1, lng1, lnb1, /*resmode=*/1, hA, nullptr, hB);

  // Layer 2
  k_zero<<<2048, 256, 0, stream>>>(sbuf, NREL * N_NODES * HDIM);
  k_scatter<HDIM><<<(N_EDGES * HDIM + 255) / 256, 256, 0, stream>>>(src, dstp, et, hB, sbuf);
  k_rgcn<HDIM, 64, 6><<<rgcnBlocks, 128, rgcnSmem, stream>>>(
      hB, sbuf, inv, r2, w2, b2, lng2, lnb2, /*resmode=*/1, hB, nullptr, hA);

  // Decoder (E % 128 == 0)
  k_decoder<<<N_EDGES / 128, 256, decSmem, stream>>>(
      src, dstp, hA, ea, dw1, db1, dw2, db2, dw3, db3, outp);
}